// GAT_MUTAG_27633819582785
// MI455X (gfx1250) — compile-verified
//
#include <hip/hip_runtime.h>
#include <hip/hip_bf16.h>
#include <math.h>
#include <limits.h>

typedef __attribute__((ext_vector_type(2))) float v2f;
typedef __attribute__((ext_vector_type(8))) float v8f;

#define N_NODES 100000
#define N_EDGES 3200000
#define N_GRAPHS 512
#define NEG_SLOPE 0.2f

// ---------- helpers: order-preserving float<->int for atomicMax ----------
__device__ __forceinline__ int f2o(float f) {
    int i = __float_as_int(f);
    return (i >= 0) ? i : (i ^ 0x7FFFFFFF);
}
__device__ __forceinline__ float o2f(int i) {
    int b = (i >= 0) ? i : (i ^ 0x7FFFFFFF);
    return __int_as_float(b);
}

// ---------- padding kernels (make GEMM operands rectangular) -------------
// x [M,14] -> xpad [M,16], zero-filled tail
__global__ void pad_x(const float* __restrict__ x, float* __restrict__ xp, int M) {
    int i = blockIdx.x * blockDim.x + threadIdx.x;
    if (i >= M * 16) return;
    int r = i >> 4, c = i & 15;
    xp[i] = (c < 14) ? x[r * 14 + c] : 0.0f;
}
// W4 [64,2] -> W4pad [64,16], zero-filled tail
__global__ void pad_w(const float* __restrict__ w, float* __restrict__ wp,
                      int K, int Nout) {
    int i = blockIdx.x * blockDim.x + threadIdx.x;
    if (i >= K * 16) return;
    int r = i >> 4, c = i & 15;
    wp[i] = (c < Nout) ? w[r * Nout + c] : 0.0f;
}

// ---------- WMMA GEMM: H[M,Nout] = A[M,K] x W[K,LDB] --------------------
// A-stationary strip GEMM: each wave owns one 16-row M-tile and sweeps its
// A fragment across NT 16-wide N-tiles (NT*K/4 chained v_wmma_f32_16x16x4_f32,
// fully unrolled). Block = 128 threads = 4 waves = 4 M-tiles.
template <int K, int LDB, int NT>
__global__ __launch_bounds__(128) void gemm_wmma(const float* __restrict__ A,
                                                 const float* __restrict__ W,
                                                 float* __restrict__ H,
                                                 int Nout, int Mtiles) {
    const int lane   = threadIdx.x & 31;
    const int wave   = threadIdx.x >> 5;
    const int mTile  = blockIdx.x * 4 + wave;
    if (mTile >= Mtiles) return;

    const int mBase  = mTile * 16;
    const int laneLo = lane & 15;
    const int laneHi = lane >> 4;                 // 0 or 1

    const float* aPtr = A + (size_t)(mBase + laneLo) * K + laneHi * 2;
    const float* bPtr = W + (size_t)(laneHi * 2) * LDB + laneLo;

    v8f acc[NT];
#pragma unroll
    for (int t = 0; t < NT; ++t) acc[t] = (v8f){};

#pragma unroll
    for (int k0 = 0; k0 < K; k0 += 4) {
        v2f a = *(const v2f*)(aPtr + k0);         // global_load_b64, aligned
#pragma unroll
        for (int t = 0; t < NT; ++t) {
            v2f b;
            b.x = bPtr[k0 * LDB + t * 16];
            b.y = bPtr[(k0 + 1) * LDB + t * 16];
            acc[t] = __builtin_amdgcn_wmma_f32_16x16x4_f32(
                /*neg_a=*/false, a, /*neg_b=*/false, b,
                /*c_mod=*/(short)0, acc[t], /*reuse_a=*/false, /*reuse_b=*/false);
        }
    }

#pragma unroll
    for (int t = 0; t < NT; ++t) {
        const int nCol = t * 16 + laneLo;
        if (nCol < Nout) {
#pragma unroll
            for (int r = 0; r < 8; ++r) {
                const int m = mBase + r + 8 * laneHi;   // C/D VGPR layout
                H[(size_t)m * Nout + nCol] = acc[t][r];
            }
        }
    }
}

// ---------- per-node attention dot products -----------------------------
__global__ void att_dots(const float* __restrict__ H,
                         const float* __restrict__ att_s,
                         const float* __restrict__ att_d,
                         float* __restrict__ as_, float* __restrict__ ad_,
                         int M, int F) {
    int i = blockIdx.x * blockDim.x + threadIdx.x;
    if (i >= M) return;
    float s = 0.f, d = 0.f;
    const float* row = H + (size_t)i * F;
    for (int c = 0; c < F; ++c) {
        float v = row[c];
        s += v * att_s[c];
        d += v * att_d[c];
    }
    as_[i] = s;
    ad_[i] = d;
}

// ---------- per-layer accumulator init ----------------------------------
__global__ void init_layer(float* __restrict__ agg, int* __restrict__ maxb,
                           float* __restrict__ denom, int MF, int M) {
    int i = blockIdx.x * blockDim.x + threadIdx.x;
    if (i < MF) agg[i] = 0.f;
    if (i < M) { maxb[i] = INT_MIN; denom[i] = 0.f; }
}

// ---------- edge pass A: segment max ------------------------------------
__global__ void edge_max(const int* __restrict__ src, const int* __restrict__ dst,
                         const float* __restrict__ as_, const float* __restrict__ ad_,
                         int* __restrict__ maxb, int E, int Et) {
    int e = blockIdx.x * blockDim.x + threadIdx.x;
    if (e >= Et) return;
    int s, d;
    if (e < E) { s = src[e]; d = dst[e]; } else { s = d = e - E; }   // self loops
    float v = as_[s] + ad_[d];
    v = (v > 0.f) ? v : v * NEG_SLOPE;
    atomicMax(&maxb[d], f2o(v));
}

// ---------- edge pass B: exp numerators + denominators ------------------
__global__ void edge_expsum(const int* __restrict__ src, const int* __restrict__ dst,
                            const float* __restrict__ as_, const float* __restrict__ ad_,
                            const int* __restrict__ maxb, float* __restrict__ denom,
                            float* __restrict__ e_num, int E, int Et) {
    int e = blockIdx.x * blockDim.x + threadIdx.x;
    if (e >= Et) return;
    int s, d;
    if (e < E) { s = src[e]; d = dst[e]; } else { s = d = e - E; }
    float v = as_[s] + ad_[d];
    v = (v > 0.f) ? v : v * NEG_SLOPE;
    float ev = expf(v - o2f(maxb[d]));
    e_num[e] = ev;
    atomicAdd(&denom[d], ev);
}

// ---------- edge pass C: one wave per edge, alpha*h[src] -> agg[dst] ----
__global__ void edge_agg(const int* __restrict__ src, const int* __restrict__ dst,
                         const float* __restrict__ e_num, const float* __restrict__ denom,
                         const float* __restrict__ H, float* __restrict__ agg,
                         int E, int Et, int F) {
    int gwave = (blockIdx.x * blockDim.x + threadIdx.x) >> 5;
    int lane  = threadIdx.x & 31;
    if (gwave >= Et) return;
    int s, d;
    if (gwave < E) { s = src[gwave]; d = dst[gwave]; } else { s = d = gwave - E; }
    float alpha = e_num[gwave] / denom[d];          // redundant per lane, L2-hot
    for (int c = lane * 2; c < F; c += 64) {
        float2 hv = *(const float2*)&H[(size_t)s * F + c];
        atomicAdd(&agg[(size_t)d * F + c],     alpha * hv.x);
        atomicAdd(&agg[(size_t)d * F + c + 1], alpha * hv.y);
    }
}

// ---------- bias + relu (in-place); F is a power of two ------------------
__global__ void bias_relu(float* __restrict__ agg, const float* __restrict__ b,
                          int MF, int F) {
    int i = blockIdx.x * blockDim.x + threadIdx.x;
    if (i >= MF) return;
    float v = agg[i] + b[i & (F - 1)];
    agg[i] = (v > 0.f) ? v : 0.f;
}

// ---------- pooling ------------------------------------------------------
__global__ void pool_init(float* __restrict__ gsum, float* __restrict__ gcnt,
                          int G, int C) {
    int i = blockIdx.x * blockDim.x + threadIdx.x;
    if (i < G * C) gsum[i] = 0.f;
    if (i < G) gcnt[i] = 0.f;
}

__global__ void pool_sum(const float* __restrict__ H, const int* __restrict__ batch,
                         float* __restrict__ gsum, float* __restrict__ gcnt,
                         int M, int C) {
    int i = blockIdx.x * blockDim.x + threadIdx.x;
    if (i >= M) return;
    int g = batch[i];
    for (int c = 0; c < C; ++c)
        atomicAdd(&gsum[(size_t)g * C + c], H[(size_t)i * C + c]);
    atomicAdd(&gcnt[g], 1.f);
}

__global__ void log_softmax(const float* __restrict__ gsum, const float* __restrict__ gcnt,
                            float* __restrict__ out, int G, int C) {
    int g = blockIdx.x * blockDim.x + threadIdx.x;
    if (g >= G) return;
    float n = fmaxf(gcnt[g], 1.f);
    float m = -INFINITY;
    for (int c = 0; c < C; ++c) m = fmaxf(m, gsum[(size_t)g * C + c] / n);
    float lse = 0.f;
    for (int c = 0; c < C; ++c) lse += expf(gsum[(size_t)g * C + c] / n - m);
    lse = m + logf(lse);
    for (int c = 0; c < C; ++c) out[(size_t)g * C + c] = gsum[(size_t)g * C + c] / n - lse;
}

// ------------------------------------------------------------------------
extern "C" void kernel_launch(void* const* d_in, const int* in_sizes, int n_in,
                              void* d_out, int out_size, void* d_ws, size_t ws_size,
                              hipStream_t stream) {
    const int M      = N_NODES;
    const int E      = N_EDGES;
    const int Et     = E + M;       // with self loops
    const int G      = N_GRAPHS;
    const int C      = 2;
    const int Mtiles = M / 16;      // 6250 (exact)
    const int gemmBlocks = (Mtiles + 3) / 4;

    const float* x     = (const float*)d_in[0];
    const int*   ei    = (const int*)d_in[1];
    const int*   batch = (const int*)d_in[2];
    const int*   src   = ei;
    const int*   dst   = ei + E;

    const float *Wl[4], *As[4], *Ad[4], *Bi[4];
    for (int l = 0; l < 4; ++l) {
        Wl[l] = (const float*)d_in[3 + 4 * l];
        As[l] = (const float*)d_in[4 + 4 * l];
        Ad[l] = (const float*)d_in[5 + 4 * l];
        Bi[l] = (const float*)d_in[6 + 4 * l];
    }

    // workspace carve-up (floats)
    float* buf0  = (float*)d_ws;                    // h  = in @ W      [M,64]
    float* buf1  = buf0 + (size_t)M * 64;           // agg / layer out  [M,64]
    float* a_s   = buf1 + (size_t)M * 64;           // [M]
    float* a_d   = a_s + M;                         // [M]
    int*   maxb  = (int*)(a_d + M);                 // [M]
    float* denom = (float*)(maxb + M);              // [M]
    float* e_num = denom + M;                       // [Et]
    float* gsum  = e_num + Et;                      // [G*C]
    float* gcnt  = gsum + (size_t)G * C;            // [G]
    float* xpad  = gcnt + G;                        // [M,16]
    float* w4pad = xpad + (size_t)M * 16;           // [64,16]

    // per-call operand padding
    pad_x<<<(M * 16 + 255) / 256, 256, 0, stream>>>(x, xpad, M);
    pad_w<<<(64 * 16 + 255) / 256, 256, 0, stream>>>(Wl[3], w4pad, 64, 2);

    const int dims[5] = {14, 64, 64, 64, 2};
    const float* in = xpad;

    for (int l = 0; l < 4; ++l) {
        const int F  = dims[l + 1];
        const int MF = M * F;

        // dense GEMM (WMMA), A-stationary 16x64 strips
        if (l == 0)
            gemm_wmma<16, 64, 4><<<gemmBlocks, 128, 0, stream>>>(in, Wl[0], buf0, 64, Mtiles);
        else if (l == 3)
            gemm_wmma<64, 16, 1><<<gemmBlocks, 128, 0, stream>>>(in, w4pad, buf0, 2, Mtiles);
        else
            gemm_wmma<64, 64, 4><<<gemmBlocks, 128, 0, stream>>>(in, Wl[l], buf0, 64, Mtiles);

        att_dots<<<(M + 255) / 256, 256, 0, stream>>>(buf0, As[l], Ad[l], a_s, a_d, M, F);
        init_layer<<<(MF + 255) / 256, 256, 0, stream>>>(buf1, maxb, denom, MF, M);
        edge_max<<<(Et + 255) / 256, 256, 0, stream>>>(src, dst, a_s, a_d, maxb, E, Et);
        edge_expsum<<<(Et + 255) / 256, 256, 0, stream>>>(src, dst, a_s, a_d, maxb, denom, e_num, E, Et);
        {
            long long threads = (long long)Et * 32;
            int blocks = (int)((threads + 255) / 256);
            edge_agg<<<blocks, 256, 0, stream>>>(src, dst, e_num, denom, buf0, buf1, E, Et, F);
        }
        bias_relu<<<(MF + 255) / 256, 256, 0, stream>>>(buf1, Bi[l], MF, F);
        in = buf1;   // layer output becomes next input
    }

    pool_init<<<(G * C + 255) / 256, 256, 0, stream>>>(gsum, gcnt, G, C);
    pool_sum<<<(M + 255) / 256, 256, 0, stream>>>(buf1, batch, gsum, gcnt, M, C);
    log_softmax<<<(G + 63) / 64, 64, 0, stream>>>(gsum, gcnt, (float*)d_out, G, C);
}